// Kmeans_21285857919395
// MI455X (gfx1250) — compile-verified
//
#include <hip/hip_runtime.h>
#include <hip/hip_bf16.h>

typedef __attribute__((ext_vector_type(2))) float v2f;
typedef __attribute__((ext_vector_type(8))) float v8f;

#define EPSF 1e-12f
#define DIM 128           // latent_dim
#define KCLUST 64         // number of clusters
#define LDSPITCH 132      // padded row pitch (dwords): banks (4n+c)%64 -> conflict-free b64 reads
#define ROWS_PER_WAVE 16
#define WAVES_PER_BLOCK 8
#define ROWS_PER_BLOCK (ROWS_PER_WAVE * WAVES_PER_BLOCK)

// One inverse-distance term: 1 / (||m - x||^p + eps).
// P2 = true is the hot path (p == 2 -> dist^p == d2, no transcendentals).
template <bool P2>
__device__ __forceinline__ float inv_term(float dot, float xx, float mm, float pf) {
    float d2 = fmaxf(xx + mm - 2.0f * dot, 0.0f);
    float dp;
    if (P2) dp = d2;
    else    dp = __powf(sqrtf(d2), pf);
    return 1.0f / (dp + EPSF);
}

// Per-wave harmonic-mean reduction over the 16x64 WMMA result.
template <bool P2>
__device__ __forceinline__ float reduce_rows(const v8f& acc0, const v8f& acc1,
                                             const v8f& acc2, const v8f& acc3,
                                             const float* xxv,
                                             float mm0, float mm1, float mm2, float mm3,
                                             int r0, int hl, int ln, int nbatch, float pf) {
    float wsum = 0.0f;
    #pragma unroll
    for (int r = 0; r < 8; ++r) {
        float s = inv_term<P2>(acc0[r], xxv[r], mm0, pf)
                + inv_term<P2>(acc1[r], xxv[r], mm1, pf)
                + inv_term<P2>(acc2[r], xxv[r], mm2, pf)
                + inv_term<P2>(acc3[r], xxv[r], mm3, pf);
        // sum the 64 cluster terms: 4 per lane above, 16 lanes of the half-wave below
        s += __shfl_xor(s, 1, 32);
        s += __shfl_xor(s, 2, 32);
        s += __shfl_xor(s, 4, 32);
        s += __shfl_xor(s, 8, 32);
        if (ln == 0) {
            int row = r0 + hl * 8 + r;
            if (row < nbatch) wsum += (float)KCLUST / (s + EPSF);
        }
    }
    return wsum;
}

__global__ void __launch_bounds__(256)
hkmeans_wmma_kernel(const float* __restrict__ X, const float* __restrict__ M,
                    const int* __restrict__ pptr, float* __restrict__ out,
                    int nbatch, float scale) {
    __shared__ float sM[KCLUST * LDSPITCH];   // 64 x 132 floats (~33 KB of the 320 KB WGP LDS)
    __shared__ float sMM[KCLUST];             // ||m_k||^2
    __shared__ float sWave[WAVES_PER_BLOCK];

    const int tid  = threadIdx.x;
    const int wave = tid >> 5;
    const int lane = tid & 31;
    const int hl   = lane >> 4;     // half-wave id (selects K-pair of the fragment)
    const int ln   = lane & 15;     // row/col within fragment

    // ---- Stage M into LDS (coalesced float4), padded pitch ----
    {
        const float4* M4 = (const float4*)M;
        for (int i = tid; i < (KCLUST * DIM) / 4; i += 256) {
            float4 v = M4[i];
            int row = i >> 5;            // (i*4)/128
            int col = (i & 31) << 2;     // (i*4)%128
            *(float4*)&sM[row * LDSPITCH + col] = v;
        }
    }
    __syncthreads();
    // ---- ||m||^2 per cluster ----
    if (tid < KCLUST) {
        float s = 0.0f;
        #pragma unroll 8
        for (int c = 0; c < DIM; ++c) {
            float v = sM[tid * LDSPITCH + c];
            s += v * v;
        }
        sMM[tid] = s;
    }
    __syncthreads();

    // ---- GEMM: 16-row X tile per wave, D = Xtile @ M^T via v_wmma_f32_16x16x4_f32 ----
    const int r0 = blockIdx.x * ROWS_PER_BLOCK + wave * ROWS_PER_WAVE;
    int rowA = r0 + ln;
    if (rowA > nbatch - 1) rowA = nbatch - 1;           // clamp (branchless, EXEC stays full)
    const float* xrow = X + (size_t)rowA * DIM + hl * 2; // A frag: lane ln holds row ln, K-pair hl*2

    v8f acc0 = {}, acc1 = {}, acc2 = {}, acc3 = {};
    float xxp = 0.0f;                                   // partial ||x||^2 over this lane's K subset

    #pragma unroll 4
    for (int k0 = 0; k0 < DIM; k0 += 4) {
        v2f a = *(const v2f*)(xrow + k0);
        xxp += a.x * a.x + a.y * a.y;

        const float* mb = &sM[ln * LDSPITCH + hl * 2 + k0];
        v2f b0 = *(const v2f*)(mb);
        v2f b1 = *(const v2f*)(mb + 16 * LDSPITCH);
        v2f b2 = *(const v2f*)(mb + 32 * LDSPITCH);
        v2f b3 = *(const v2f*)(mb + 48 * LDSPITCH);

        acc0 = __builtin_amdgcn_wmma_f32_16x16x4_f32(false, a, false, b0, (short)0, acc0, false, false);
        acc1 = __builtin_amdgcn_wmma_f32_16x16x4_f32(false, a, false, b1, (short)0, acc1, false, false);
        acc2 = __builtin_amdgcn_wmma_f32_16x16x4_f32(false, a, false, b2, (short)0, acc2, false, false);
        acc3 = __builtin_amdgcn_wmma_f32_16x16x4_f32(false, a, false, b3, (short)0, acc3, false, false);
    }

    // ---- ||x||^2 per row: lanes l and l+16 hold the two K-halves of row l&15 ----
    xxp += __shfl_xor(xxp, 16, 32);
    float xxv[8];
    #pragma unroll
    for (int r = 0; r < 8; ++r)
        xxv[r] = __shfl(xxp, hl * 8 + r, 32);   // row served by C/D vgpr r in this half-wave

    const float mm0 = sMM[ln];
    const float mm1 = sMM[16 + ln];
    const float mm2 = sMM[32 + ln];
    const float mm3 = sMM[48 + ln];
    const int p = *pptr;                         // scalar (uniform) load
    const float pf = (float)p;

    // ---- Harmonic-mean reduction; uniform scalar branch keeps powf off the p==2 path ----
    float wsum;
    if (p == 2) {
        wsum = reduce_rows<true >(acc0, acc1, acc2, acc3, xxv, mm0, mm1, mm2, mm3,
                                  r0, hl, ln, nbatch, pf);
    } else {
        wsum = reduce_rows<false>(acc0, acc1, acc2, acc3, xxv, mm0, mm1, mm2, mm3,
                                  r0, hl, ln, nbatch, pf);
    }

    wsum += __shfl_xor(wsum, 16, 32);
    if (lane == 0) sWave[wave] = wsum;
    __syncthreads();

    if (tid == 0) {
        float t = 0.0f;
        #pragma unroll
        for (int w = 0; w < WAVES_PER_BLOCK; ++w) t += sWave[w];
        atomicAdd(out, t * scale);
    }
}

extern "C" void kernel_launch(void* const* d_in, const int* in_sizes, int n_in,
                              void* d_out, int out_size, void* d_ws, size_t ws_size,
                              hipStream_t stream) {
    const float* X = (const float*)d_in[0];
    const float* M = (const float*)d_in[1];
    const int*   p = (const int*)d_in[2];
    float* out = (float*)d_out;

    const int nbatch = in_sizes[0] / DIM;
    if (nbatch <= 0) return;

    // loss normalizer: 1 / (nbatch * K * latent_dim)
    const float scale = 1.0f / ((float)nbatch * (float)KCLUST * (float)DIM);

    hipMemsetAsync(d_out, 0, (size_t)out_size * sizeof(float), stream);

    const int grid = (nbatch + ROWS_PER_BLOCK - 1) / ROWS_PER_BLOCK;
    hkmeans_wmma_kernel<<<grid, 256, 0, stream>>>(X, M, p, out, nbatch, scale);
}